// Dense3DSpatialTransformer_77584289234968
// MI455X (gfx1250) — compile-verified
//
#include <hip/hip_runtime.h>

// Dense 3D spatial transformer (trilinear warp with implicit zero padding).
// Bandwidth-bound streaming gather: NT b128 flow loads / NT b128 output
// stores keep the 192MB L2 free to hold the whole 55MB image for gathers.
//
// Launch geometry carries the coordinate decomposition so no per-thread
// integer div/mod chains are needed:
//   blockDim = (56, 4, 1)  -> zg = tid.x (56 quads * 4 voxels = D=224)
//                             xc = blockIdx.x*4 + tid.y
//   gridDim  = (48, 160, 2)-> yc = blockIdx.y (SALU), bc = blockIdx.z (SALU)
// 224 threads/block = exactly 7 wave32s, all dims divide exactly.

typedef float __attribute__((ext_vector_type(4))) f32x4;

constexpr int Bc = 2, Hc = 160, Wc = 192, Dc = 224;

__device__ __forceinline__ int iclamp(int v, int lo, int hi) {
    return v < lo ? lo : (v > hi ? hi : v);
}

__global__ __launch_bounds__(224)
void Dense3DSpatialTransformer_kernel(const float* __restrict__ I,
                                      const float* __restrict__ flow,
                                      float* __restrict__ out) {
    const int zg = threadIdx.x;                       // 0..55
    const int xc = blockIdx.x * 4 + threadIdx.y;      // 0..191
    const int yc = blockIdx.y;                        // 0..159  (uniform)
    const int bc = blockIdx.z;                        // 0..1    (uniform)
    const int z4 = zg * 4;

    // uniform (SGPR) bases
    const float* __restrict__ Ib = I + (size_t)bc * ((size_t)Hc * Wc * Dc);
    const size_t rowbase = (((size_t)bc * Hc + yc) * Wc) * Dc;  // uniform
    const size_t voxbase = rowbase + (size_t)xc * Dc + z4;      // per-lane

    // Prefetch the identity-position gather neighborhood (flow displacements
    // are ~ +/-1 voxel) while the flow loads are still in flight.
    __builtin_prefetch(Ib + ((size_t)yc * Wc + xc) * Dc + z4, 0, 0);

    // 4 voxels * (dx,dy,dz) interleaved = 12 floats = 3 aligned b128 NT loads
    const f32x4* __restrict__ fp = (const f32x4*)(flow + voxbase * 3);
    const f32x4 f0 = __builtin_nontemporal_load(fp + 0);
    const f32x4 f1 = __builtin_nontemporal_load(fp + 1);
    const f32x4 f2 = __builtin_nontemporal_load(fp + 2);

    const float dxs[4] = {f0.x, f0.w, f1.z, f2.y};
    const float dys[4] = {f0.y, f1.x, f1.w, f2.z};
    const float dzs[4] = {f0.z, f1.y, f2.x, f2.w};

    const float xf = (float)xc + 1.0f;   // padded-volume coordinate bases
    const float yf = (float)yc + 1.0f;
    const float zf = (float)z4 + 1.0f;

    f32x4 res;
#pragma unroll
    for (int k = 0; k < 4; ++k) {
        // sample coords in the padded volume (pad shifts everything by +1)
        const float xs = dxs[k] + xf;
        const float ys = dys[k] + yf;
        const float zs = dzs[k] + (zf + (float)k);

        const int fx = (int)floorf(xs);
        const int fy = (int)floorf(ys);
        const int fz = (int)floorf(zs);

        // clipped padded-volume corner indices (padded max = dim + 1)
        const int x0 = iclamp(fx,     0, Wc + 1), x1 = iclamp(fx + 1, 0, Wc + 1);
        const int y0 = iclamp(fy,     0, Hc + 1), y1 = iclamp(fy + 1, 0, Hc + 1);
        const int z0 = iclamp(fz,     0, Dc + 1), z1 = iclamp(fz + 1, 0, Dc + 1);

        // reference computes weights from the *clipped* upper corner
        const float dxw = (float)x1 - xs;
        const float dyw = (float)y1 - ys;
        const float dzw = (float)z1 - zs;

        // unpadded indices; pad border (index -1 or dim) gathers zero -> fold
        // the zero into the per-axis weight factor instead (branchless, EXEC
        // stays full so the 8 gathers remain plain clause-able b32 loads).
        const int xi0 = x0 - 1, xi1 = x1 - 1;
        const int yi0 = y0 - 1, yi1 = y1 - 1;
        const int zi0 = z0 - 1, zi1 = z1 - 1;

        const bool vx0 = (unsigned)xi0 < (unsigned)Wc, vx1 = (unsigned)xi1 < (unsigned)Wc;
        const bool vy0 = (unsigned)yi0 < (unsigned)Hc, vy1 = (unsigned)yi1 < (unsigned)Hc;
        const bool vz0 = (unsigned)zi0 < (unsigned)Dc, vz1 = (unsigned)zi1 < (unsigned)Dc;

        const float ax0 = vx0 ? dxw : 0.0f, ax1 = vx1 ? (1.0f - dxw) : 0.0f;
        const float ay0 = vy0 ? dyw : 0.0f, ay1 = vy1 ? (1.0f - dyw) : 0.0f;
        const float az0 = vz0 ? dzw : 0.0f, az1 = vz1 ? (1.0f - dzw) : 0.0f;

        const int cx0 = vx0 ? xi0 : 0, cx1 = vx1 ? xi1 : 0;
        const int cy0 = vy0 ? yi0 : 0, cy1 = vy1 ? yi1 : 0;
        const int cz0 = vz0 ? zi0 : 0, cz1 = vz1 ? zi1 : 0;

        const int ry0 = cy0 * (Wc * Dc), ry1 = cy1 * (Wc * Dc);
        const int rx0 = cx0 * Dc,        rx1 = cx1 * Dc;

        // 8 corner gathers (regular temporal hint: image stays L2-resident)
        const float v000 = Ib[ry0 + rx0 + cz0];  // (y0,x0,z0)
        const float v100 = Ib[ry1 + rx0 + cz0];  // (y1,x0,z0)
        const float v010 = Ib[ry0 + rx1 + cz0];  // (y0,x1,z0)
        const float v110 = Ib[ry1 + rx1 + cz0];  // (y1,x1,z0)
        const float v001 = Ib[ry0 + rx0 + cz1];  // (y0,x0,z1)
        const float v101 = Ib[ry1 + rx0 + cz1];  // (y1,x0,z1)
        const float v011 = Ib[ry0 + rx1 + cz1];  // (y0,x1,z1)
        const float v111 = Ib[ry1 + rx1 + cz1];  // (y1,x1,z1)

        res[k] = az0 * (ax0 * (ay0 * v000 + ay1 * v100) +
                        ax1 * (ay0 * v010 + ay1 * v110)) +
                 az1 * (ax0 * (ay0 * v001 + ay1 * v101) +
                        ax1 * (ay0 * v011 + ay1 * v111));
    }

    // write-once stream: NT b128 store
    __builtin_nontemporal_store(res, (f32x4*)(out + voxbase));
}

extern "C" void kernel_launch(void* const* d_in, const int* in_sizes, int n_in,
                              void* d_out, int out_size, void* d_ws, size_t ws_size,
                              hipStream_t stream) {
    const float* I    = (const float*)d_in[0];   // [2,160,192,224,1] f32
    const float* flow = (const float*)d_in[1];   // [2,160,192,224,3] f32
    float* out        = (float*)d_out;           // [2,160,192,224,1] f32

    dim3 block(56, 4, 1);                        // 224 threads = 7 wave32s
    dim3 grid(Wc / 4, Hc, Bc);                   // (48, 160, 2)
    Dense3DSpatialTransformer_kernel<<<grid, block, 0, stream>>>(I, flow, out);
}